// TransformerGraphDecoder_9783935500472
// MI455X (gfx1250) — compile-verified
//
#include <hip/hip_runtime.h>
#include <hip/hip_bf16.h>
#include <math.h>

// ---------------------------------------------------------------------------
// TransformerConv (PyG-style) for MI455X / gfx1250.
//   N=50000 nodes, E=800000 edges, IN=128, H=4 heads, C=32 (H*C=128).
// GEMMs use V_WMMA_F32_16X16X4_F32 (fp32-exact matrix path, wave32).
// Edge phase uses coalesced float4 gathers + native f32 atomics.
// ---------------------------------------------------------------------------

typedef __attribute__((ext_vector_type(2))) float v2f;
typedef __attribute__((ext_vector_type(8))) float v8f;

namespace cfg {
constexpr int N  = 50000;
constexpr int E  = 800000;
constexpr int IN = 128;
constexpr int H  = 4;
constexpr int C  = 32;
constexpr int HC = 128;       // H*C
}

// Monotonic float <-> unsigned mapping so we can use hardware atomicMax(u32)
// for the scatter segment-max. 0u is below every encoded float.
__device__ __forceinline__ unsigned f2ord(float f) {
    unsigned b = __float_as_uint(f);
    return (b & 0x80000000u) ? ~b : (b | 0x80000000u);
}
__device__ __forceinline__ float ord2f(unsigned o) {
    unsigned b = (o & 0x80000000u) ? (o & 0x7FFFFFFFu) : ~o;
    return __uint_as_float(b);
}

// ---------------------------------------------------------------------------
// Kernel 0: zero the per-(node,head) softmax state (max + denom).
// ---------------------------------------------------------------------------
__global__ void init_state_kernel(unsigned* __restrict__ m,
                                  float* __restrict__ denom) {
    int i = blockIdx.x * blockDim.x + threadIdx.x;
    if (i < cfg::N * cfg::H) {
        m[i] = 0u;          // below encode(-inf)
        denom[i] = 0.0f;
    }
}

// ---------------------------------------------------------------------------
// Kernel 1: fused q/k/v/skip projection.
//   out512 = x[50000,128] @ [Wq|Wk|Wv|Wskip][128,512] + biases
// Block = 256 threads = 8 waves, one 16-row stripe per block (50000 = 3125*16).
// Wave w owns 64 columns of the concatenated output:
//   waves 0-1 -> q, 2-3 -> k, 4-5 -> v, 6-7 -> skip (written to d_out).
// Inner loop: K in steps of 4 with V_WMMA_F32_16X16X4_F32.
//   A frag (16x4 f32): lane 0-15 -> M=lane, {v0,v1}={K,K+1}; lanes 16-31 K+=2.
//   B frag (4x16 f32): VGPR j -> row K=kk+j+2*hi, lane -> N (row striped
//   across lanes within one VGPR).
// ---------------------------------------------------------------------------
__global__ __launch_bounds__(256)
void qkvs_gemm_kernel(const float* __restrict__ x,
                      const float* __restrict__ Wq, const float* __restrict__ bq,
                      const float* __restrict__ Wk, const float* __restrict__ bk,
                      const float* __restrict__ Wv, const float* __restrict__ bv,
                      const float* __restrict__ Wsk, const float* __restrict__ bsk,
                      float* __restrict__ qo, float* __restrict__ ko,
                      float* __restrict__ vo, float* __restrict__ so) {
    const int lane = threadIdx.x & 31;
    const int wav  = threadIdx.x >> 5;          // 0..7
    const int m0   = blockIdx.x * 16;           // row stripe
    const int lo   = lane & 15;
    const int hi   = lane >> 4;

    const float* Wm;
    const float* Bm;
    float*       Om;
    switch (wav >> 1) {
        case 0:  Wm = Wq;  Bm = bq;  Om = qo; break;
        case 1:  Wm = Wk;  Bm = bk;  Om = ko; break;
        case 2:  Wm = Wv;  Bm = bv;  Om = vo; break;
        default: Wm = Wsk; Bm = bsk; Om = so; break;
    }
    const int col0 = (wav & 1) * 64;            // 4 tiles of 16 cols

    v8f acc[4] = {};                            // 4 x (16x16 f32) accumulators

    const float2* x2 = reinterpret_cast<const float2*>(x);
    const int arow = m0 + lo;

    for (int kk = 0; kk < cfg::IN; kk += 4) {
        // A fragment: x[arow, kk+2*hi .. kk+2*hi+1]
        float2 av = x2[arow * (cfg::IN / 2) + (kk >> 1) + hi];
        v2f a; a.x = av.x; a.y = av.y;

        const int krow = (kk + 2 * hi) * cfg::HC;
#pragma unroll
        for (int t = 0; t < 4; ++t) {
            const int col = col0 + t * 16 + lo;
            v2f b;
            b.x = Wm[krow + col];               // K = kk + 2*hi
            b.y = Wm[krow + cfg::HC + col];     // K = kk + 2*hi + 1
            acc[t] = __builtin_amdgcn_wmma_f32_16x16x4_f32(
                false, a, false, b, (short)0, acc[t], false, false);
        }
    }

    // Epilogue: bias add + store. C/D layout: VGPR r, lanes 0-15 -> M=r,
    // lanes 16-31 -> M=r+8; N = lane%16.
#pragma unroll
    for (int t = 0; t < 4; ++t) {
        const int col  = col0 + t * 16 + lo;
        const float bb = Bm[col];
#pragma unroll
        for (int r = 0; r < 8; ++r) {
            const int row = m0 + r + 8 * hi;
            Om[row * cfg::HC + col] = acc[t][r] + bb;
        }
    }
}

// ---------------------------------------------------------------------------
// Kernel 2: per-edge attention logits + scatter-max.
// One wave32 per edge: lane loads float4 of q[dst] and k[src] (512B/row,
// fully coalesced), xor-shuffle reduce within 8-lane head groups.
// ---------------------------------------------------------------------------
__global__ __launch_bounds__(256)
void edge_logits_kernel(const float* __restrict__ q,
                        const float* __restrict__ k,
                        const long long* __restrict__ ei,
                        float* __restrict__ alpha,
                        unsigned* __restrict__ m) {
    const int lane = threadIdx.x & 31;
    const int wav  = threadIdx.x >> 5;
    const int e = blockIdx.x * 8 + wav;
    if (e >= cfg::E) return;

    const int src = (int)ei[e];
    const int dst = (int)ei[cfg::E + e];

    const float4 qv = reinterpret_cast<const float4*>(q)[dst * 32 + lane];
    const float4 kv = reinterpret_cast<const float4*>(k)[src * 32 + lane];
    float p = qv.x * kv.x + qv.y * kv.y + qv.z * kv.z + qv.w * kv.w;

    // reduce over the 8 lanes of each head (lane>>3 == head)
    p += __shfl_xor(p, 1, 32);
    p += __shfl_xor(p, 2, 32);
    p += __shfl_xor(p, 4, 32);

    if ((lane & 7) == 0) {
        const int h = lane >> 3;
        const float a = p * 0.17677669529663688f;   // 1/sqrt(C=32)
        alpha[e * cfg::H + h] = a;
        atomicMax(&m[dst * cfg::H + h], f2ord(a));
    }
}

// ---------------------------------------------------------------------------
// Kernel 3: ex = exp(alpha - max[dst]); scatter-add denominators.
// One thread per (edge, head).
// ---------------------------------------------------------------------------
__global__ __launch_bounds__(256)
void edge_exp_kernel(const long long* __restrict__ ei,
                     const unsigned* __restrict__ m,
                     float* __restrict__ alpha,
                     float* __restrict__ denom) {
    const int idx = blockIdx.x * blockDim.x + threadIdx.x;
    if (idx >= cfg::E * cfg::H) return;
    const int e = idx >> 2;      // H == 4
    const int h = idx & 3;
    const int dst = (int)ei[cfg::E + e];
    const float mx = ord2f(m[dst * cfg::H + h]);
    const float ex = __expf(alpha[idx] - mx);
    alpha[idx] = ex;
    __hip_atomic_fetch_add(&denom[dst * cfg::H + h], ex,
                           __ATOMIC_RELAXED, __HIP_MEMORY_SCOPE_AGENT);
}

// ---------------------------------------------------------------------------
// Kernel 4: weighted scatter of values: out[dst] += v[src] * w[e,h].
// One wave32 per edge; coalesced float4 gather of v[src]; native f32 atomics.
// d_out already holds the skip projection from kernel 1.
// ---------------------------------------------------------------------------
__global__ __launch_bounds__(256)
void edge_scatter_kernel(const float* __restrict__ v,
                         const long long* __restrict__ ei,
                         const float* __restrict__ ex,
                         const float* __restrict__ denom,
                         float* __restrict__ out) {
    const int lane = threadIdx.x & 31;
    const int wav  = threadIdx.x >> 5;
    const int e = blockIdx.x * 8 + wav;
    if (e >= cfg::E) return;

    const int src = (int)ei[e];
    const int dst = (int)ei[cfg::E + e];
    const int h = lane >> 3;

    const float wgt = ex[e * cfg::H + h] / denom[dst * cfg::H + h];
    const float4 vv = reinterpret_cast<const float4*>(v)[src * 32 + lane];

    float* op = out + (size_t)dst * cfg::HC + lane * 4;
    __hip_atomic_fetch_add(op + 0, vv.x * wgt, __ATOMIC_RELAXED, __HIP_MEMORY_SCOPE_AGENT);
    __hip_atomic_fetch_add(op + 1, vv.y * wgt, __ATOMIC_RELAXED, __HIP_MEMORY_SCOPE_AGENT);
    __hip_atomic_fetch_add(op + 2, vv.z * wgt, __ATOMIC_RELAXED, __HIP_MEMORY_SCOPE_AGENT);
    __hip_atomic_fetch_add(op + 3, vv.w * wgt, __ATOMIC_RELAXED, __HIP_MEMORY_SCOPE_AGENT);
}

// ---------------------------------------------------------------------------
// Host-side launcher.
// ---------------------------------------------------------------------------
extern "C" void kernel_launch(void* const* d_in, const int* in_sizes, int n_in,
                              void* d_out, int out_size, void* d_ws, size_t ws_size,
                              hipStream_t stream) {
    using namespace cfg;
    const float*      x   = (const float*)d_in[0];
    const long long*  ei  = (const long long*)d_in[1];   // [2, E] int64
    const float*      Wq  = (const float*)d_in[2];
    const float*      bq  = (const float*)d_in[3];
    const float*      Wk  = (const float*)d_in[4];
    const float*      bk  = (const float*)d_in[5];
    const float*      Wv  = (const float*)d_in[6];
    const float*      bv  = (const float*)d_in[7];
    const float*      Wsk = (const float*)d_in[8];
    const float*      bsk = (const float*)d_in[9];
    float* out = (float*)d_out;

    // Workspace carve-up (~91 MB):
    float* q     = (float*)d_ws;                       // N*HC
    float* k     = q + (size_t)N * HC;                 // N*HC
    float* v     = k + (size_t)N * HC;                 // N*HC
    float* alpha = v + (size_t)N * HC;                 // E*H (logits -> exp)
    unsigned* m  = (unsigned*)(alpha + (size_t)E * H); // N*H ordered-uint max
    float* denom = (float*)(m + (size_t)N * H);        // N*H

    // 0) reset softmax state (must be re-zeroed every call)
    {
        const int n = N * H;
        init_state_kernel<<<(n + 255) / 256, 256, 0, stream>>>(m, denom);
    }
    // 1) fused projections: q,k,v -> ws ; skip(+bias) -> d_out
    qkvs_gemm_kernel<<<N / 16, 256, 0, stream>>>(
        x, Wq, bq, Wk, bk, Wv, bv, Wsk, bsk, q, k, v, out);
    // 2) logits + scatter max
    edge_logits_kernel<<<E / 8, 256, 0, stream>>>(q, k, ei, alpha, m);
    // 3) exp + denom
    edge_exp_kernel<<<(E * H + 255) / 256, 256, 0, stream>>>(ei, m, alpha, denom);
    // 4) weighted scatter-add into out (already holds skip)
    edge_scatter_kernel<<<E / 8, 256, 0, stream>>>(v, ei, alpha, denom, out);
}